// CALayer_36567351558175
// MI455X (gfx1250) — compile-verified
//
#include <hip/hip_runtime.h>
#include <math.h>

// ---------------------------------------------------------------------------
// MI455X (gfx1250) CALayer kernel, round 4.
//   * 13->512->12 folded into one 12x13 weight (exact algebra).
//   * Conv1d(26->24) as WMMA GEMM with bias folded in as augmented K-column.
//   * Attention: head-padded projections (j'=4h+d, Q pre-scaled 1/sqrt(3)),
//     per-head score GEMM (K=4), float4 register softmax, AV GEMM against a
//     transpose-stored V, output in-place into spare columns of score tiles.
//   * One wave32 per batch element; zero-padded LDS tiles; all fragment
//     accesses unconditional; compile-time strides (contiguous -> b64).
//   * Round-4 fix: no "+0" bias add on bias-less GEMM writebacks.
// ---------------------------------------------------------------------------

typedef float v2f __attribute__((ext_vector_type(2)));
typedef float v8f __attribute__((ext_vector_type(8)));

#if defined(__gfx1250__) && __has_builtin(__builtin_amdgcn_wmma_f32_16x16x4_f32)
#define USE_WMMA_F32X4 1
#else
#define USE_WMMA_F32X4 0
#endif

// ---- packed (pre-padded) weight table in d_ws (float offsets) -------------
// Invariant: every [16][16] weight block is followed by its padded-16 bias.
#define OFF_WF     0      // fused co_k∘pre_lin [16][16] (12x13 real)
#define OFF_BF     256
#define OFF_COQW   272    // co_q_w [16][16]
#define OFF_COQB   528
#define OFF_COLW   544    // co_lin_w [16][32] (k-fast)
#define OFF_COLB   1056
#define OFF_C1QW   1072   // head-padded, pre-scaled 1/sqrt(3)
#define OFF_C1KW   1344   // head-padded
#define OFF_C1VW   1616   // head-padded
#define OFF_C2QW   1888   // head-padded, pre-scaled
#define OFF_C2KW   2160
#define OFF_C2VW   2432
#define OFF_PCWA   2704   // pre_conv augmented [32][32]: [w | b | 0...]
#define OFF_FCW    3728   // 24
#define OFF_FCB    3752   // 1
#define OFF_OLW    3753   // 6x24
#define OFF_OLB    3897   // 6
#define OFF_LNG    3903
#define OFF_LNB    3904
#define WTOT       3908

// ---- per-batch LDS tiles (float offsets after weights) --------------------
#define D_P    0      // [32][16] normalized prompt (24x12 real)
#define D_PP   512    // [32][16] normalized preds
#define D_ENT  1024   // [16][32] enc transposed (+bias col 26); dead after E1
#define D_E1   1536   // [32][16] conv out; dead after kp
#define D_VT   1024   // [32][32] transposed V, ALIASES ENT+E1 (rows16+ zeroed)
#define D_QP   2048   // [32][16]
#define D_KP   2560   // [32][16]
#define D_CO   3072   // [32][16] coat
#define D_QB   3584   // [32][16] Q proj (head-padded)
#define D_KB   4096   // [32][16] K proj (head-padded)
#define D_ST   4608   // 4 x [32][32]: qk/aq/pc/ck, reused as per-head scores
#define D_QK   (D_ST)
#define D_AQ   (D_ST + 1024)
#define D_PC   (D_ST + 2048)
#define D_CK   (D_ST + 3072)
#define D_CA1  8704   // 24x12 compact
#define D_CA2  8992
#define D_FUS  9280
#define DTOT   9312

#define LDS_FLOATS (WTOT + DTOT)   // 13220 floats = 52.9 KB

// ---------------------------------------------------------------------------
// Small GEMM via V_WMMA_F32_16X16X4_F32. Compile-time strides, no branches.
// All A fragments preloaded before any store (enables in-place C=A-tile use).
// Caller guarantees operand tiles valid/zero over the padded index ranges.
// ---------------------------------------------------------------------------
template <int M, int N, int K, bool HAS_BIAS,
          int ASR, int ASC, int BSR, int BSC, int CSR, int CSC>
__device__ __forceinline__ void gemm_wmma(
    const float* __restrict__ A, const float* __restrict__ B,
    float* __restrict__ C, const float* __restrict__ bias, int lane)
{
  constexpr int MT = (M + 15) / 16;
  constexpr int NT = (N + 15) / 16;
  constexpr int KT = (K + 3) / 4;
  static_assert(!HAS_BIAS || NT == 1, "bias path assumes single N tile");
  const int l15 = lane & 15;
  const int hi  = lane >> 4;
  v2f af[MT][KT];
#pragma unroll
  for (int mt = 0; mt < MT; ++mt)
#pragma unroll
    for (int kt = 0; kt < KT; ++kt) {
      const int m = mt * 16 + l15, k0 = kt * 4 + hi * 2;
      if constexpr (ASC == 1) {
        af[mt][kt] = *(const v2f*)(A + m * ASR + k0);
      } else {
        af[mt][kt].x = A[m * ASR + (k0    ) * ASC];
        af[mt][kt].y = A[m * ASR + (k0 + 1) * ASC];
      }
    }
  float bv = 0.f;
  if constexpr (HAS_BIAS) bv = bias[l15];
#pragma unroll
  for (int nt = 0; nt < NT; ++nt) {
    const int n = nt * 16 + l15;
    v2f bf[KT];
#pragma unroll
    for (int kt = 0; kt < KT; ++kt) {
      const int k0 = kt * 4 + hi * 2;
      if constexpr (BSR == 1) {
        bf[kt] = *(const v2f*)(B + k0 + n * BSC);
      } else {
        bf[kt].x = B[(k0    ) * BSR + n * BSC];
        bf[kt].y = B[(k0 + 1) * BSR + n * BSC];
      }
    }
#pragma unroll
    for (int mt = 0; mt < MT; ++mt) {
      v8f acc = {0.f, 0.f, 0.f, 0.f, 0.f, 0.f, 0.f, 0.f};
#pragma unroll
      for (int kt = 0; kt < KT; ++kt) {
#if USE_WMMA_F32X4
        acc = __builtin_amdgcn_wmma_f32_16x16x4_f32(
            false, af[mt][kt], false, bf[kt], (short)0, acc, false, false);
#else
        acc[0] += af[mt][kt].x * bf[kt].x + af[mt][kt].y * bf[kt].y; // host
#endif
      }
#pragma unroll
      for (int r = 0; r < 8; ++r) {
        const int ci = (mt * 16 + r + hi * 8) * CSR + n * CSC;
        if constexpr (HAS_BIAS) C[ci] = acc[r] + bv;   // pk-add pairs
        else                    C[ci] = acc[r];        // no "+0" VALU cost
      }
    }
  }
}

// ---------------------------------------------------------------------------
// Full cross-attention (H=4, hd=3, L=24) using WMMA throughout.
// Weight blocks are head-padded [16][16] with bias at +256 (Q pre-scaled).
// Uses D_QB, D_KB, D_VT and the 4 score tiles at D_ST; writes compact out.
// ---------------------------------------------------------------------------
__device__ __forceinline__ void cross_attention(
    float* __restrict__ D, const float* __restrict__ Wl,
    int WQ, int WK, int WV, int qsrc, int kvsrc, int resid, int outOff,
    int lane)
{
  gemm_wmma<24, 16, 12, true, 16, 1, 1, 16, 16, 1>(
      D + qsrc, Wl + WQ, D + D_QB, Wl + WQ + 256, lane);
  gemm_wmma<24, 16, 12, true, 16, 1, 1, 16, 16, 1>(
      D + kvsrc, Wl + WK, D + D_KB, Wl + WK + 256, lane);
  gemm_wmma<24, 16, 12, true, 16, 1, 1, 16, 1, 32>(      // transposed store
      D + kvsrc, Wl + WV, D + D_VT, Wl + WV + 256, lane);
  __syncthreads();
  // Per-head scores (already scaled via Q weights), full-tile writes.
#pragma unroll
  for (int h = 0; h < 4; ++h)
    gemm_wmma<24, 24, 4, false, 16, 1, 1, 16, 32, 1>(
        D + D_QB + 4 * h, D + D_KB + 4 * h, D + D_ST + h * 1024, nullptr, lane);
  __syncthreads();
  // Softmax over k: 96 rows (4 heads x 24 q) spread over 32 lanes.
#pragma unroll
  for (int i = 0; i < 3; ++i) {
    const int r = i * 32 + lane;        // 0..95
    const int h = r / 24, q = r - h * 24;
    float* row = D + D_ST + h * 1024 + q * 32;
    float s[24];
    const float4* rp = (const float4*)row;
#pragma unroll
    for (int t4 = 0; t4 < 6; ++t4) {
      const float4 v = rp[t4];
      s[4*t4+0] = v.x; s[4*t4+1] = v.y; s[4*t4+2] = v.z; s[4*t4+3] = v.w;
    }
    float mx = s[0];
#pragma unroll
    for (int k = 1; k < 24; ++k) mx = fmaxf(mx, s[k]);
    float sum = 0.f;
#pragma unroll
    for (int k = 0; k < 24; ++k) { s[k] = __expf(s[k] - mx); sum += s[k]; }
    const float inv = 1.f / sum;
    float4* wp = (float4*)row;
#pragma unroll
    for (int t4 = 0; t4 < 6; ++t4) {
      float4 v;
      v.x = s[4*t4+0]*inv; v.y = s[4*t4+1]*inv;
      v.z = s[4*t4+2]*inv; v.w = s[4*t4+3]*inv;
      wp[t4] = v;
    }
  }
  __syncthreads();
  // AV per head: in-place into cols 16..31 of the same score tile (safe: all
  // A fragments are preloaded before any store inside gemm_wmma).
#pragma unroll
  for (int h = 0; h < 4; ++h)
    gemm_wmma<24, 16, 24, false, 32, 1, 1, 32, 32, 1>(
        D + D_ST + h * 1024, D + D_VT + 4 * h * 32,
        D + D_ST + h * 1024 + 16, nullptr, lane);
  __syncthreads();
  // Combine heads + residual into compact [24][12].
  if (lane < 24) {
#pragma unroll
    for (int j = 0; j < 12; ++j) {
      const int h = j / 3, dd = j - h * 3;
      D[outOff + lane * 12 + j] =
          D[resid + lane * 16 + j] + D[D_ST + h * 1024 + lane * 32 + 16 + dd];
    }
  }
}

// ---------------------------------------------------------------------------
// Kernel 1: fold 13->512->12, pack every weight pre-padded / head-padded.
// ---------------------------------------------------------------------------
__global__ __launch_bounds__(256) void fuse_pack_kernel(
    const float* __restrict__ pre_conv_w, const float* __restrict__ pre_conv_b,
    const float* __restrict__ pre_lin_w,  const float* __restrict__ pre_lin_b,
    const float* __restrict__ co_q_w,     const float* __restrict__ co_q_b,
    const float* __restrict__ co_k_w,     const float* __restrict__ co_k_b,
    const float* __restrict__ co_lin_w,   const float* __restrict__ co_lin_b,
    const float* __restrict__ c1qw, const float* __restrict__ c1qb,
    const float* __restrict__ c1kw, const float* __restrict__ c1kb,
    const float* __restrict__ c1vw, const float* __restrict__ c1vb,
    const float* __restrict__ c2qw, const float* __restrict__ c2qb,
    const float* __restrict__ c2kw, const float* __restrict__ c2kb,
    const float* __restrict__ c2vw, const float* __restrict__ c2vb,
    const float* __restrict__ fcw,  const float* __restrict__ fcb,
    const float* __restrict__ olw,  const float* __restrict__ olb,
    const float* __restrict__ lng,  const float* __restrict__ lnb,
    float* __restrict__ W)
{
  const int t = threadIdx.x;
  const float SC = 0.5773502691896258f;  // 1/sqrt(3)
  // Fused weight Wf[j,d] = sum_m co_k_w[j,m]*pre_lin_w[m,d], padded [16][16].
  {
    const int j = t >> 4, d = t & 15;
    float s = 0.f;
    if (j < 12 && d < 13)
      for (int m = 0; m < 512; ++m) s += co_k_w[j * 512 + m] * pre_lin_w[m * 13 + d];
    W[OFF_WF + t] = s;
  }
  if (t < 16) {
    float s = 0.f;
    if (t < 12) {
      s = co_k_b[t];
      for (int m = 0; m < 512; ++m) s += co_k_w[t * 512 + m] * pre_lin_b[m];
    }
    W[OFF_BF + t] = s;
  }
  // co_q: plain padded [16][16] + bias.
  {
    const int r = t >> 4, c = t & 15;
    W[OFF_COQW + t] = (r < 12 && c < 12) ? co_q_w[r * 12 + c] : 0.f;
  }
  if (t < 16) W[OFF_COQB + t] = (t < 12) ? co_q_b[t] : 0.f;
  // co_lin_w padded [16][32] (k-fast) + bias.
  for (int i = t; i < 512; i += 256) {
    const int r = i >> 5, c = i & 31;
    W[OFF_COLW + i] = (r < 12 && c < 24) ? co_lin_w[r * 24 + c] : 0.f;
  }
  if (t < 16) W[OFF_COLB + t] = (t < 12) ? co_lin_b[t] : 0.f;
  // Head-padded attention weights: out channel j' = 4h+dd <- j = 3h+dd.
#define PADHEAD(dst, wsrc, bsrc, scale)                                       \
  { const int r = t >> 4, c = t & 15;                                         \
    const int h = r >> 2, dd = r & 3;                                         \
    W[(dst) + t] = (dd < 3 && c < 12) ? (wsrc)[(h*3+dd)*12 + c] * (scale) : 0.f; } \
  if (t < 16) { const int h2 = t >> 2, d2 = t & 3;                            \
    W[(dst) + 256 + t] = (d2 < 3) ? (bsrc)[h2*3+d2] * (scale) : 0.f; }
  PADHEAD(OFF_C1QW, c1qw, c1qb, SC);
  PADHEAD(OFF_C1KW, c1kw, c1kb, 1.f);
  PADHEAD(OFF_C1VW, c1vw, c1vb, 1.f);
  PADHEAD(OFF_C2QW, c2qw, c2qb, SC);
  PADHEAD(OFF_C2KW, c2kw, c2kb, 1.f);
  PADHEAD(OFF_C2VW, c2vw, c2vb, 1.f);
#undef PADHEAD
  // pre_conv augmented [32][32]: cols 0..25 = w, col 26 = bias, rest 0.
  for (int i = t; i < 1024; i += 256) {
    const int r = i >> 5, c = i & 31;
    float v = 0.f;
    if (r < 24) {
      if (c < 26) v = pre_conv_w[r * 26 + c];
      else if (c == 26) v = pre_conv_b[r];
    }
    W[OFF_PCWA + i] = v;
  }
  // Compact tails.
  if (t < 24) W[OFF_FCW + t] = fcw[t];
  if (t == 0) W[OFF_FCB] = fcb[0];
  for (int i = t; i < 144; i += 256) W[OFF_OLW + i] = olw[i];
  if (t < 6)  W[OFF_OLB + t] = olb[t];
  if (t == 0) { W[OFF_LNG] = lng[0]; W[OFF_LNB] = lnb[0]; }
  for (int i = 3905 + t; i < WTOT; i += 256) W[i] = 0.f;
}

// ---------------------------------------------------------------------------
// Kernel 2: one wave32 per batch element.
// ---------------------------------------------------------------------------
__global__ __launch_bounds__(32) void ca_layer_kernel(
    const float* __restrict__ prompt, const float* __restrict__ preds,
    const float* __restrict__ enc, const float* __restrict__ W,
    float* __restrict__ out)
{
  __shared__ float lds[LDS_FLOATS];
  const int lane = threadIdx.x;
  const int b = blockIdx.x;
  float* Wl = lds;
  float* D  = lds + WTOT;

  // Weights -> LDS (float4).
  {
    const float4* src = (const float4*)W;
    float4* dst = (float4*)Wl;
    for (int i = lane; i < WTOT / 4; i += 32) dst[i] = src[i];
  }
  // Zero VALU-written padded tiles: P,PP,ENT [0,1536) and AQ,PC [D_AQ,D_CK).
  {
    const float4 z = make_float4(0.f, 0.f, 0.f, 0.f);
    float4* z0 = (float4*)(D);
    for (int i = lane; i < 384; i += 32) z0[i] = z;
    float4* z1 = (float4*)(D + D_AQ);
    for (int i = lane; i < 512; i += 32) z1[i] = z;
  }
  // Inputs.
  const float* pg = prompt + (size_t)b * 288;
  const float* qg = preds  + (size_t)b * 288;
  const float* eg = enc    + (size_t)b * 338;
  if (lane < 24) {
    const float4* s0 = (const float4*)(pg + lane * 12);
    float4* t0 = (float4*)(D + D_P + lane * 16);
    t0[0] = s0[0]; t0[1] = s0[1]; t0[2] = s0[2];
    const float4* s1 = (const float4*)(qg + lane * 12);
    float4* t1 = (float4*)(D + D_PP + lane * 16);
    t1[0] = s1[0]; t1[1] = s1[1]; t1[2] = s1[2];
  }
  for (int i = lane; i < 338; i += 32) {        // enc transposed into ENT
    const int c = i / 13, d = i - c * 13;
    D[D_ENT + d * 32 + c] = eg[i];
  }
  __syncthreads();

  // L2-normalize p, pp rows; enc columns (= ENT rows) + bias column.
  if (lane < 24) {
    float s = 0.f;
#pragma unroll
    for (int d = 0; d < 12; ++d) { float v = D[D_P + lane * 16 + d]; s += v * v; }
    float inv = 1.f / fmaxf(sqrtf(s), 1e-12f);
#pragma unroll
    for (int d = 0; d < 12; ++d) D[D_P + lane * 16 + d] *= inv;
    s = 0.f;
#pragma unroll
    for (int d = 0; d < 12; ++d) { float v = D[D_PP + lane * 16 + d]; s += v * v; }
    inv = 1.f / fmaxf(sqrtf(s), 1e-12f);
#pragma unroll
    for (int d = 0; d < 12; ++d) D[D_PP + lane * 16 + d] *= inv;
  }
  if (lane < 13) {
    float s = 0.f;
#pragma unroll
    for (int c = 0; c < 26; ++c) { float v = D[D_ENT + lane * 32 + c]; s += v * v; }
    const float inv = 1.f / fmaxf(sqrtf(s), 1e-12f);
#pragma unroll
    for (int c = 0; c < 26; ++c) D[D_ENT + lane * 32 + c] *= inv;
    D[D_ENT + lane * 32 + 26] = 1.0f;          // bias column
  }
  __syncthreads();

  // E1 = pre_conv(en) + b  (bias folded as K=27 column)      [24][13]
  gemm_wmma<24, 13, 27, false, 32, 1, 1, 32, 16, 1>(
      Wl + OFF_PCWA, D + D_ENT, D + D_E1, nullptr, lane);
  __syncthreads();
  // kp = E1 @ Wf^T + bf                                      [24][12]
  gemm_wmma<24, 12, 13, true, 16, 1, 1, 16, 16, 1>(
      D + D_E1, Wl + OFF_WF, D + D_KP, Wl + OFF_BF, lane);
  // qp = p @ co_q_w^T + co_q_b                               [24][12]
  gemm_wmma<24, 12, 12, true, 16, 1, 1, 16, 16, 1>(
      D + D_P, Wl + OFF_COQW, D + D_QP, Wl + OFF_COQB, lane);
  __syncthreads();
  // qk = qp @ kp^T                                           [24][24]
  gemm_wmma<24, 24, 12, false, 16, 1, 1, 16, 32, 1>(
      D + D_QP, D + D_KP, D + D_QK, nullptr, lane);
  // Zero VT upper half (old E1 region) for later AV reads (rows 16..31).
  {
    const float4 z = make_float4(0.f, 0.f, 0.f, 0.f);
    float4* z2 = (float4*)(D + D_E1);
    for (int i = lane; i < 128; i += 32) z2[i] = z;
  }
  __syncthreads();

  // a_q: softmax over q (columns of qk) -> AQ, in registers.
  if (lane < 24) {
    float e[24];
    float mx = -3.0e38f;
#pragma unroll
    for (int q = 0; q < 24; ++q) { e[q] = D[D_QK + q * 32 + lane]; mx = fmaxf(mx, e[q]); }
    float sum = 0.f;
#pragma unroll
    for (int q = 0; q < 24; ++q) { e[q] = __expf(e[q] - mx); sum += e[q]; }
    const float inv = 1.f / sum;
#pragma unroll
    for (int q = 0; q < 24; ++q) D[D_AQ + q * 32 + lane] = e[q] * inv;
  }
  __syncthreads();
  // a_k: softmax over k (rows of qk) in place, float4 row access.
  if (lane < 24) {
    float* row = D + D_QK + lane * 32;
    float s[24];
    const float4* rp = (const float4*)row;
#pragma unroll
    for (int t4 = 0; t4 < 6; ++t4) {
      const float4 v = rp[t4];
      s[4*t4+0] = v.x; s[4*t4+1] = v.y; s[4*t4+2] = v.z; s[4*t4+3] = v.w;
    }
    float mx = s[0];
#pragma unroll
    for (int k = 1; k < 24; ++k) mx = fmaxf(mx, s[k]);
    float sum = 0.f;
#pragma unroll
    for (int k = 0; k < 24; ++k) { s[k] = __expf(s[k] - mx); sum += s[k]; }
    const float inv = 1.f / sum;
    float4* wp = (float4*)row;
#pragma unroll
    for (int t4 = 0; t4 < 6; ++t4) {
      float4 v;
      v.x = s[4*t4+0]*inv; v.y = s[4*t4+1]*inv;
      v.z = s[4*t4+2]*inv; v.w = s[4*t4+3]*inv;
      wp[t4] = v;
    }
    // pc[:, 0:12] = p
    const float4* sp = (const float4*)(D + D_P + lane * 16);
    float4* dp = (float4*)(D + D_PC + lane * 32);
    dp[0] = sp[0]; dp[1] = sp[1]; dp[2] = sp[2];
  }
  __syncthreads();
  // c_q = a_q @ kp -> pc[:, 12:24]                           [24][12]
  gemm_wmma<24, 12, 24, false, 32, 1, 16, 1, 32, 1>(
      D + D_AQ, D + D_KP, D + D_PC + 12, nullptr, lane);
  __syncthreads();
  // c_k[k,d] = sum_q a_k[q,k]*pc[q,d]                        [24][24]
  gemm_wmma<24, 24, 24, false, 1, 32, 32, 1, 32, 1>(
      D + D_QK, D + D_PC, D + D_CK, nullptr, lane);
  __syncthreads();
  // coat[d,o] = sum_k c_k[k,d]*co_lin_w[o,k] + co_lin_b[o]   [24][12]
  gemm_wmma<24, 12, 24, true, 1, 32, 1, 32, 16, 1>(
      D + D_CK, Wl + OFF_COLW, D + D_CO, Wl + OFF_COLB, lane);
  __syncthreads();

  // Cross-attentions (WMMA path; score tiles reuse the 4 [32][32] tiles).
  cross_attention(D, Wl, OFF_C1QW, OFF_C1KW, OFF_C1VW,
                  D_PP, D_CO, D_PP, D_CA1, lane);
  __syncthreads();
  cross_attention(D, Wl, OFF_C2QW, OFF_C2KW, OFF_C2VW,
                  D_CO, D_PP, D_CO, D_CA2, lane);
  __syncthreads();

  // fus[l] = leaky_relu( sum_c fcw[c]*concat(ca1,ca2)[c,l] + fcb )
  if (lane < 24) {
    float s = Wl[OFF_FCB];
#pragma unroll
    for (int c = 0; c < 24; ++c) {
      const float x = (lane < 12) ? D[D_CA1 + c * 12 + lane]
                                  : D[D_CA2 + c * 12 + (lane - 12)];
      s += Wl[OFF_FCW + c] * x;
    }
    D[D_FUS + lane] = (s > 0.f) ? s : 0.01f * s;
  }
  __syncthreads();

  // out_lin + LayerNorm over singleton last axis.
  if (lane < 6) {
    float s = Wl[OFF_OLB + lane];
#pragma unroll
    for (int l = 0; l < 24; ++l) s += Wl[OFF_OLW + lane * 24 + l] * D[D_FUS + l];
    const float mu = s;
    const float dv = s - mu;
    const float var = dv * dv;
    out[(size_t)b * 6 + lane] = dv * rsqrtf(var + 1e-5f) * Wl[OFF_LNG] + Wl[OFF_LNB];
  }
}

// ---------------------------------------------------------------------------
extern "C" void kernel_launch(void* const* d_in, const int* in_sizes, int n_in,
                              void* d_out, int out_size, void* d_ws, size_t ws_size,
                              hipStream_t stream) {
  const float* prompt = (const float*)d_in[0];
  const float* preds  = (const float*)d_in[1];
  const float* enc    = (const float*)d_in[2];
  const float* pre_conv_w = (const float*)d_in[3];
  const float* pre_conv_b = (const float*)d_in[4];
  const float* pre_lin_w  = (const float*)d_in[5];
  const float* pre_lin_b  = (const float*)d_in[6];
  const float* co_q_w  = (const float*)d_in[7];
  const float* co_q_b  = (const float*)d_in[8];
  const float* co_k_w  = (const float*)d_in[9];
  const float* co_k_b  = (const float*)d_in[10];
  const float* co_lin_w = (const float*)d_in[11];
  const float* co_lin_b = (const float*)d_in[12];
  const float* c1qw = (const float*)d_in[13]; const float* c1qb = (const float*)d_in[14];
  const float* c1kw = (const float*)d_in[15]; const float* c1kb = (const float*)d_in[16];
  const float* c1vw = (const float*)d_in[17]; const float* c1vb = (const float*)d_in[18];
  const float* c2qw = (const float*)d_in[19]; const float* c2qb = (const float*)d_in[20];
  const float* c2kw = (const float*)d_in[21]; const float* c2kb = (const float*)d_in[22];
  const float* c2vw = (const float*)d_in[23]; const float* c2vb = (const float*)d_in[24];
  const float* fcw = (const float*)d_in[25]; const float* fcb = (const float*)d_in[26];
  const float* olw = (const float*)d_in[27]; const float* olb = (const float*)d_in[28];
  const float* lng = (const float*)d_in[29]; const float* lnb = (const float*)d_in[30];

  float* W = (float*)d_ws;
  const int B = in_sizes[0] / (24 * 12);

  fuse_pack_kernel<<<1, 256, 0, stream>>>(
      pre_conv_w, pre_conv_b, pre_lin_w, pre_lin_b,
      co_q_w, co_q_b, co_k_w, co_k_b, co_lin_w, co_lin_b,
      c1qw, c1qb, c1kw, c1kb, c1vw, c1vb,
      c2qw, c2qb, c2kw, c2kb, c2vw, c2vb,
      fcw, fcb, olw, olb, lng, lnb, W);

  ca_layer_kernel<<<dim3(B), dim3(32), 0, stream>>>(
      prompt, preds, enc, W, (float*)d_out);
}